// CorrelationGNN_29686813950696
// MI455X (gfx1250) — compile-verified
//
#include <hip/hip_runtime.h>

typedef __attribute__((ext_vector_type(2))) float v2f;
typedef __attribute__((ext_vector_type(8))) float v8f;

#define THREADS 256

// ---------- degree / normalization ----------
__global__ __launch_bounds__(THREADS) void k_deg_init(float* deg, int n) {
  int i = blockIdx.x * THREADS + threadIdx.x;
  if (i < n) deg[i] = 1.0f;  // self-loop contribution
}

__global__ __launch_bounds__(THREADS) void k_deg_accum(float* deg,
                                                       const int* __restrict__ dst,
                                                       int e) {
  int i = blockIdx.x * THREADS + threadIdx.x;
  if (i < e) atomicAdd(&deg[dst[i]], 1.0f);
}

__global__ __launch_bounds__(THREADS) void k_rsqrt_inplace(float* d, int n) {
  int i = blockIdx.x * THREADS + threadIdx.x;
  if (i < n) d[i] = rsqrtf(d[i]);  // deg >= 1 always (self loops)
}

// ---------- GEMM: C[n,32] = A[n,32] @ W[32,32] (+bias), fp32 WMMA 16x16x4 ----------
// One wave computes one 16-row tile, all 32 output columns (two 16x16 accumulators).
// A-frag (16x4, MxK): lanes 0-15 -> K = k0+{0,1}; lanes 16-31 -> K = k0+{2,3}; M = lane&15.
// B-frag (4x16, KxN): N = lane&15; K striped like A. C/D: VGPR r -> M = r + 8*(lane>>4), N = lane&15.
__global__ __launch_bounds__(THREADS) void k_gemm32(const float* A,
                                                    const float* __restrict__ W,
                                                    float* C,
                                                    const float* __restrict__ bias,
                                                    int n) {
  const int lane = threadIdx.x & 31;
  const int wave = threadIdx.x >> 5;
  const long tile = (long)blockIdx.x * 8 + wave;  // 16-row tile index
  const long row0 = tile * 16;
  if (row0 >= n) return;  // wave-uniform: EXEC stays all-1s for WMMA

  const int l15   = lane & 15;
  const int khalf = (lane >> 4) * 2;  // 0 or 2

  v8f c0 = {};  // cols 0..15
  v8f c1 = {};  // cols 16..31

#pragma unroll
  for (int k0 = 0; k0 < 32; k0 += 4) {
    const float* ap = A + (row0 + l15) * 32 + k0 + khalf;
    v2f a;
    a.x = ap[0];
    a.y = ap[1];

    const float* bp = W + (k0 + khalf) * 32 + l15;
    v2f b0;
    b0.x = bp[0];
    b0.y = bp[32];
    v2f b1;
    b1.x = bp[16];
    b1.y = bp[48];

    c0 = __builtin_amdgcn_wmma_f32_16x16x4_f32(false, a, false, b0, (short)0, c0, false, false);
    c1 = __builtin_amdgcn_wmma_f32_16x16x4_f32(false, a, false, b1, (short)0, c1, false, false);
  }

  const int rbase = (lane >> 4) * 8;
  float badd0 = 0.0f, badd1 = 0.0f;
  if (bias) {
    badd0 = bias[l15];
    badd1 = bias[16 + l15];
  }
#pragma unroll
  for (int r = 0; r < 8; ++r) {
    const long row = row0 + rbase + r;
    C[row * 32 + l15]      = c0[r] + badd0;
    C[row * 32 + 16 + l15] = c1[r] + badd1;
  }
}

// ---------- self-loop term: agg[i,:] = T[i,:] * dinv[i]^2 ----------
__global__ __launch_bounds__(THREADS) void k_self_init(const float* __restrict__ T,
                                                       const float* __restrict__ dinv,
                                                       float* __restrict__ agg,
                                                       long total) {
  long i = (long)blockIdx.x * THREADS + threadIdx.x;
  if (i < total) {
    float dv = dinv[i >> 5];
    agg[i] = T[i] * dv * dv;
  }
}

// ---------- edge scatter: agg[dst,:] += T[src,:] * dinv[src]*dinv[dst] ----------
// One wave per edge, lane = feature column (32 cols -> fully coalesced 128B atomics).
__global__ __launch_bounds__(THREADS) void k_edge_scatter(const float* __restrict__ T,
                                                          float* agg,
                                                          const int* __restrict__ src,
                                                          const int* __restrict__ dst,
                                                          const float* __restrict__ dinv,
                                                          int nEdges) {
  const int lane = threadIdx.x & 31;
  const long e = (long)blockIdx.x * 8 + (threadIdx.x >> 5);
  if (e >= nEdges) return;
  const int s = src[e];
  const int d = dst[e];
  const float coef = dinv[s] * dinv[d];
  const float val = T[(long)s * 32 + lane] * coef;
  atomicAdd(&agg[(long)d * 32 + lane], val);
}

// ---------- h = relu(agg + b) in place ----------
__global__ __launch_bounds__(THREADS) void k_bias_relu(float* __restrict__ h,
                                                       const float* __restrict__ bias,
                                                       long total) {
  long i = (long)blockIdx.x * THREADS + threadIdx.x;
  if (i < total) {
    float v = h[i] + bias[i & 31];
    h[i] = v > 0.0f ? v : 0.0f;
  }
}

extern "C" void kernel_launch(void* const* d_in, const int* in_sizes, int n_in,
                              void* d_out, int out_size, void* d_ws, size_t ws_size,
                              hipStream_t stream) {
  const float* x  = (const float*)d_in[0];
  const int*   ei = (const int*)d_in[1];
  const float* W0 = (const float*)d_in[2];
  const float* b0 = (const float*)d_in[3];
  const float* W1 = (const float*)d_in[4];
  const float* b1 = (const float*)d_in[5];
  const float* Wf = (const float*)d_in[6];
  const float* bf = (const float*)d_in[7];
  float* out = (float*)d_out;

  const int n = in_sizes[0] / 32;   // 100000 nodes
  const int E = in_sizes[1] / 2;    // 1600000 edges
  const int* src = ei;
  const int* dst = ei + E;

  // workspace layout: dinv [n floats], S [n*32 floats]
  float* dinv = (float*)d_ws;
  size_t off = ((size_t)n * sizeof(float) + 511) & ~(size_t)511;
  float* S = (float*)((char*)d_ws + off);

  const long total = (long)n * 32;
  const int gN   = (n + THREADS - 1) / THREADS;
  const int gE   = (E + THREADS - 1) / THREADS;
  const int gE8  = (E + 7) / 8;                       // one wave per edge
  const int gT   = (int)((total + THREADS - 1) / THREADS);
  const int gG   = (n / 16 + 7) / 8;                  // gemm: 8 tiles per block

  // degree -> dinv
  k_deg_init<<<gN, THREADS, 0, stream>>>(dinv, n);
  k_deg_accum<<<gE, THREADS, 0, stream>>>(dinv, dst, E);
  k_rsqrt_inplace<<<gN, THREADS, 0, stream>>>(dinv, n);

  // ---- layer 0: T0 = x@W0 (S); agg0 (out); h0 = relu(agg0+b0) (out) ----
  k_gemm32<<<gG, THREADS, 0, stream>>>(x, W0, S, nullptr, n);
  k_self_init<<<gT, THREADS, 0, stream>>>(S, dinv, out, total);
  k_edge_scatter<<<gE8, THREADS, 0, stream>>>(S, out, src, dst, dinv, E);
  k_bias_relu<<<gT, THREADS, 0, stream>>>(out, b0, total);

  // ---- layer 1: T1 = h0@W1 (S); agg1 (out); h1 = relu(agg1+b1) (out) ----
  k_gemm32<<<gG, THREADS, 0, stream>>>(out, W1, S, nullptr, n);
  k_self_init<<<gT, THREADS, 0, stream>>>(S, dinv, out, total);
  k_edge_scatter<<<gE8, THREADS, 0, stream>>>(S, out, src, dst, dinv, E);
  k_bias_relu<<<gT, THREADS, 0, stream>>>(out, b1, total);

  // ---- final: out = h1@Wf + bf (in place; each wave reads only rows it writes) ----
  k_gemm32<<<gG, THREADS, 0, stream>>>(out, Wf, out, bf, n);
}